// InterNodeMoELayerIn_2199023256087
// MI455X (gfx1250) — compile-verified
//
#include <hip/hip_runtime.h>
#include <hip/hip_bf16.h>

// MoE top-1 router + capacity scatter for MI455X (gfx1250, wave32).
//
// Roofline: N=16384 tokens, D=4096, E=8, cap=2048.
//   pass1 read x: 256MB, pass3 read x (<=256MB) + write out 256MB
//   => ~768MB @ 23.3TB/s ~= 33us. Compute (1 GFLOP router GEMM) is free.
// Router GEMM runs on the matrix pipe via V_WMMA_F32_16X16X4_F32 (f32 in/out,
// matching the f32 reference exactly).

typedef __attribute__((ext_vector_type(2))) float v2f;
typedef __attribute__((ext_vector_type(4))) float v4f;   // native vector for NT builtins
typedef __attribute__((ext_vector_type(8))) float v8f;

#define NEXP_MAX 8

// ---------------------------------------------------------------------------
// Kernel 1: logits via WMMA f32 16x16x4, fused argmax -> routes[N]
// One wave (32 lanes) computes a 16-token x 16-expert logits tile (experts
// padded 8->16 with zero weights and -3e38 bias).
//
// ISA VGPR layouts (CDNA5 ISA):
//  A 16x4 f32 : lane L holds M=L%16, K = 2*(L/16) + {vgpr0, vgpr1}
//  B 4x16 f32 : lane L holds N=L%16, K = 2*(L/16) + {vgpr0, vgpr1}
//  C/D 16x16  : lane L, vgpr v -> M = v + 8*(L/16), N = L%16
// ---------------------------------------------------------------------------
__global__ void moe_router_wmma(const float* __restrict__ x,
                                const float* __restrict__ W,
                                const float* __restrict__ bias,
                                int* __restrict__ routes,
                                int N, int D, int E) {
  const int lane = threadIdx.x & 31;
  const int half = lane >> 4;   // 0 or 1
  const int sub  = lane & 15;   // A row / B col (expert) within half
  const int wavesPerBlock = blockDim.x >> 5;
  const int tile = blockIdx.x * wavesPerBlock + (threadIdx.x >> 5);
  const int row0 = tile * 16;
  if (row0 >= N) return;  // uniform per wave

  // Per-lane streaming pointers (float2 per WMMA step).
  const float* arow = x + (size_t)(row0 + sub) * D + 2 * half;
  const int esafe = (sub < E) ? sub : 0;
  const float* brow = W + (size_t)esafe * D + 2 * half;
  const bool evalid = (sub < E);

  v8f acc = {};
#pragma unroll 8
  for (int k = 0; k < D; k += 4) {
    const v2f a2 = *(const v2f*)(arow + k);
    v2f b2 = *(const v2f*)(brow + k);
    if (!evalid) { b2[0] = 0.0f; b2[1] = 0.0f; }
    // D = A(16x4) * B(4x16) + C ; f32 throughout.
    acc = __builtin_amdgcn_wmma_f32_16x16x4_f32(
        /*neg_a=*/false, a2, /*neg_b=*/false, b2,
        /*c_mod=*/(short)0, acc, /*reuse_a=*/false, /*reuse_b=*/false);
  }

  // Bias (padded experts get -3e38 so they never win argmax).
  const float blane = evalid ? bias[sub] : -3.0e38f;

  // Fused argmax across the 16 expert lanes of each half-wave.
  // xor masks 8,4,2,1 never cross the 16-lane boundary on wave32.
#pragma unroll
  for (int v = 0; v < 8; ++v) {
    float best = acc[v] + blane;
    int bidx = sub;
#pragma unroll
    for (int m = 8; m >= 1; m >>= 1) {
      const float ov = __shfl_xor(best, m);
      const int oi = __shfl_xor(bidx, m);
      // first-occurrence tie-break to match jnp.argmax
      if (ov > best || (ov == best && oi < bidx)) { best = ov; bidx = oi; }
    }
    if (sub == 0) {
      const int tok = row0 + v + 8 * half;  // C-layout row for this vgpr/half
      if (tok < N) routes[tok] = bidx;
    }
  }
}

// ---------------------------------------------------------------------------
// Kernel 2: single-block stable prefix positions + inverse slot map.
// slot_map[e*cap + p] = token index (or -1). Inverse map lets pass 3 be a
// fully coalesced gather that writes the 256MB output exactly once.
// ---------------------------------------------------------------------------
__global__ __launch_bounds__(256)
void moe_scan_assign(const int* __restrict__ routes,
                     int* __restrict__ slot_map,
                     int N, int E, int cap) {
  constexpr int NT = 256;
  __shared__ int cnt[NT][NEXP_MAX];  // 8KB of 320KB LDS
  const int t = threadIdx.x;
  const int per = (N + NT - 1) / NT;
  const int begin = t * per;
  const int end = (begin + per < N) ? (begin + per) : N;

  // Init inverse map (ordering vs. later writes guaranteed by the barriers
  // of the scan below).
  const int slots = E * cap;
  for (int i = t; i < slots; i += NT) slot_map[i] = -1;

  int local[NEXP_MAX];
#pragma unroll
  for (int e = 0; e < NEXP_MAX; ++e) local[e] = 0;
  for (int i = begin; i < end; ++i) local[routes[i]]++;
#pragma unroll
  for (int e = 0; e < NEXP_MAX; ++e) cnt[t][e] = local[e];
  __syncthreads();

  // Hillis-Steele inclusive scan over 256 threads (8 steps), 8-wide vectors.
  for (int off = 1; off < NT; off <<= 1) {
    int add[NEXP_MAX];
#pragma unroll
    for (int e = 0; e < NEXP_MAX; ++e) add[e] = (t >= off) ? cnt[t - off][e] : 0;
    __syncthreads();
#pragma unroll
    for (int e = 0; e < NEXP_MAX; ++e) cnt[t][e] += add[e];
    __syncthreads();
  }

  // Exclusive base for this thread, then stable in-order assignment.
  int pos[NEXP_MAX];
#pragma unroll
  for (int e = 0; e < NEXP_MAX; ++e) pos[e] = (t > 0) ? cnt[t - 1][e] : 0;
  for (int i = begin; i < end; ++i) {
    const int e = routes[i];
    const int p = pos[e]++;
    if (p < cap) slot_map[e * cap + p] = i;
  }
}

// ---------------------------------------------------------------------------
// Kernel 3: gather output rows. One block per output row; 128-bit moves,
// non-temporal stores (write-once 256MB stream -> keep L2 for x).
// ---------------------------------------------------------------------------
__global__ __launch_bounds__(256)
void moe_gather_rows(const float* __restrict__ x,
                     const int* __restrict__ slot_map,
                     float* __restrict__ out, int D) {
  const int row = blockIdx.x;
  const int tok = slot_map[row];  // scalarized (uniform) load
  v4f* __restrict__ dst = (v4f*)(out + (size_t)row * D);
  const int nv = D >> 2;
  if (tok >= 0) {
    const v4f* __restrict__ src = (const v4f*)(x + (size_t)tok * D);
    for (int i = threadIdx.x; i < nv; i += blockDim.x) {
      const v4f val = src[i];
      __builtin_nontemporal_store(val, &dst[i]);
    }
  } else {
    const v4f z = {0.f, 0.f, 0.f, 0.f};
    for (int i = threadIdx.x; i < nv; i += blockDim.x)
      __builtin_nontemporal_store(z, &dst[i]);
  }
}

extern "C" void kernel_launch(void* const* d_in, const int* in_sizes, int n_in,
                              void* d_out, int out_size, void* d_ws, size_t ws_size,
                              hipStream_t stream) {
  const float* x = (const float*)d_in[0];
  const float* W = (const float*)d_in[1];
  const float* b = (const float*)d_in[2];
  float* out = (float*)d_out;

  const int E = in_sizes[2];             // 8
  const int D = in_sizes[1] / E;         // 4096
  const int N = in_sizes[0] / D;         // 16384
  const int cap = out_size / (E * D);    // 2048

  int* routes = (int*)d_ws;              // N ints
  int* slot_map = routes + N;            // E*cap ints

  // K1: WMMA router + fused argmax. 8 waves/block, one 16-token tile/wave.
  const int tiles = (N + 15) / 16;
  const int wavesPerBlock = 8;
  dim3 grid1((tiles + wavesPerBlock - 1) / wavesPerBlock);
  dim3 block1(32 * wavesPerBlock);
  moe_router_wmma<<<grid1, block1, 0, stream>>>(x, W, b, routes, N, D, E);

  // K2: stable positions + inverse slot map (single block).
  moe_scan_assign<<<1, 256, 0, stream>>>(routes, slot_map, N, E, cap);

  // K3: gather rows into the [E*cap, D] output.
  moe_gather_rows<<<E * cap, 256, 0, stream>>>(x, slot_map, out, D);
}